// MembraneLayer_45380624450078
// MI455X (gfx1250) — compile-verified
//
#include <hip/hip_runtime.h>

typedef __attribute__((ext_vector_type(2))) float v2f;
typedef __attribute__((ext_vector_type(4))) float v4f;
typedef __attribute__((ext_vector_type(8))) float v8f;
typedef __attribute__((ext_vector_type(4))) int   v4i;

typedef __attribute__((address_space(1))) v4i gv4i;   // global int4
typedef __attribute__((address_space(3))) v4i lv4i;   // LDS int4

#define B_   128
#define T_   512
#define I_   700
#define O_   128
#define CH   16          // timesteps per chunk (WMMA M)
#define LDSK 708         // padded LDS row stride in floats (708 mod 64 = 4 -> conflict-free A reads)
#define NC   (T_ / CH)   // 32 chunks

#define HAS_ASYNC_LDS __has_builtin(__builtin_amdgcn_global_load_async_to_lds_b128)

__device__ __forceinline__ void wait_async0() {
#if __has_builtin(__builtin_amdgcn_s_wait_asynccnt)
    __builtin_amdgcn_s_wait_asynccnt(0);
#else
    asm volatile("s_wait_asynccnt 0x0" ::: "memory");
#endif
}

// Stage inp[b, t0:t0+16, 0:700] into one LDS buffer (16 rows, padded stride).
// Async path: global->LDS DMA, 16B per lane per op, tracked by ASYNCcnt.
__device__ __forceinline__ void stage_chunk(const float* __restrict__ inb, int t0,
                                            float* __restrict__ buf, int tid) {
    for (int idx = tid; idx < CH * (I_ / 4); idx += 256) {
        const int r  = idx / (I_ / 4);
        const int c4 = idx - r * (I_ / 4);
        const float* g = inb + (size_t)(t0 + r) * I_ + c4 * 4;
        float*       l = buf + r * LDSK + c4 * 4;
#if HAS_ASYNC_LDS
        __builtin_amdgcn_global_load_async_to_lds_b128(
            (gv4i*)const_cast<float*>(g),
            (lv4i*)l,
            /*offset=*/0, /*cpol=*/0);
#else
        *(v4f*)l = *(const v4f*)g;
#endif
    }
}

__launch_bounds__(256, 1)
__global__ void snn_fused_kernel(const float* __restrict__ inp,
                                 const float* __restrict__ w,
                                 const float* __restrict__ alpha,
                                 const float* __restrict__ beta,
                                 float* __restrict__ syn_rec,
                                 float* __restrict__ mem_rec)
{
    __shared__ float lds[2][CH * LDSK];   // 2 x 45.3 KB of the 320 KB WGP LDS

    const int b    = blockIdx.x;
    const int tid  = threadIdx.x;
    const int lane = tid & 31;
    const int wid  = tid >> 5;         // wave id 0..7 (wave32)
    const int o0   = wid * 16;         // this wave's 16 output neurons
    const int nn   = lane & 15;        // neuron within tile (replicated in both half-waves)
    const bool lo  = lane < 16;

    const float al  = alpha[o0 + nn];
    const float bt  = beta [o0 + nn];
    const float omb = 1.0f - bt;

    // t = 0 outputs are zero
    if (lo) {
        syn_rec[((size_t)b * T_ + 0) * O_ + o0 + nn] = 0.0f;
        mem_rec[((size_t)b * T_ + 0) * O_ + o0 + nn] = 0.0f;
    }

    float syn = 0.0f, mem = 0.0f;

    const float* __restrict__ inb = inp + (size_t)b * T_ * I_;   // contiguous (T, I) slab

    // Prime the pipeline: issue chunk 0 into buffer 0.
    stage_chunk(inb, 0, &lds[0][0], tid);

    for (int c = 0; c < NC; ++c) {
        const int t0  = c * CH;
        const int cur = c & 1;
        const int nxt = cur ^ 1;

#if HAS_ASYNC_LDS
        wait_async0();        // my async DMAs into lds[cur] have landed
#endif
        __syncthreads();      // everyone's DMAs landed; previous compute on lds[nxt] done

        // Overlap: kick off chunk c+1 into the other buffer while we compute on lds[cur].
        if (c + 1 < NC) {
            stage_chunk(inb, t0 + CH, &lds[nxt][0], tid);
        }

        // ---- GEMM: C(16x16) += A(16x700) x W(700x16), fp32 WMMA, K-chunks of 4 ----
        // A layout: v0 = {K=k0 (lanes 0-15), K=k0+2 (lanes 16-31)}, v1 = {k0+1, k0+3}
        const float* __restrict__ abuf = &lds[cur][0];
        v8f cacc = {0.f, 0.f, 0.f, 0.f, 0.f, 0.f, 0.f, 0.f};
        const int mrow = lane & 15;
        const int koff = (lane >> 4) * 2;   // 0 for lanes 0-15, 2 for lanes 16-31
        #pragma unroll 5
        for (int kc = 0; kc < 175; ++kc) {
            const int k = kc * 4 + koff;
            v2f av = *(const v2f*)(&abuf[mrow * LDSK + k]);       // ds_load_b64
            v2f bv;
            bv.x = w[(size_t)(k + 0) * O_ + o0 + nn];             // w rows stay L0/L2-resident
            bv.y = w[(size_t)(k + 1) * O_ + o0 + nn];
            cacc = __builtin_amdgcn_wmma_f32_16x16x4_f32(
                    /*neg_a=*/false, av, /*neg_b=*/false, bv,
                    /*c_mod=*/(short)0, cacc, /*reuse_a=*/false, /*reuse_b=*/false);
        }

        // ---- in-register scan over the 16 timesteps of this chunk ----
        // C layout: lanes 0-15 hold h[m=v][n=lane], lanes 16-31 hold h[m=v+8][n=lane-16]
        v8f so = {0.f,0.f,0.f,0.f,0.f,0.f,0.f,0.f};
        v8f mo = {0.f,0.f,0.f,0.f,0.f,0.f,0.f,0.f};
        #pragma unroll
        for (int mm = 0; mm < 16; ++mm) {
            const int v = mm & 7;
            float x = cacc[v];
            float y = __shfl_xor(x, 16, 32);           // swap half-waves
            float hm = (mm < 8) ? (lo ? x : y) : (lo ? y : x);
            float new_mem = bt * mem + omb * syn;      // uses OLD syn
            float new_syn = al * syn + hm;
            syn = new_syn;
            mem = new_mem;
            bool keep = (mm < 8) ? lo : !lo;           // pack back into C layout
            so[v] = keep ? new_syn : so[v];
            mo[v] = keep ? new_mem : mo[v];
        }

        // ---- store syn/mem for timesteps t0+1 .. t0+16 (skip t==T) ----
        const int mbase = lo ? 0 : 8;
        #pragma unroll
        for (int v = 0; v < 8; ++v) {
            const int tstep = t0 + 1 + mbase + v;
            if (tstep < T_) {
                const size_t off = ((size_t)b * T_ + tstep) * O_ + o0 + nn;
                syn_rec[off] = so[v];
                mem_rec[off] = mo[v];
            }
        }
    }
}

extern "C" void kernel_launch(void* const* d_in, const int* in_sizes, int n_in,
                              void* d_out, int out_size, void* d_ws, size_t ws_size,
                              hipStream_t stream) {
    const float* inp   = (const float*)d_in[0];   // (B, T, I) fp32
    const float* w     = (const float*)d_in[1];   // (I, O) fp32
    const float* alpha = (const float*)d_in[2];   // (1, O) fp32
    const float* beta  = (const float*)d_in[3];   // (1, O) fp32
    float* out     = (float*)d_out;
    float* syn_rec = out;                                  // (B, T, O)
    float* mem_rec = out + (size_t)B_ * T_ * O_;           // (B, T, O)

    hipLaunchKernelGGL(snn_fused_kernel, dim3(B_), dim3(256), 0, stream,
                       inp, w, alpha, beta, syn_rec, mem_rec);
}